// GNNBlock_15324443312752
// MI455X (gfx1250) — compile-verified
//
#include <hip/hip_runtime.h>
#include <hip/hip_bf16.h>

typedef float v2f __attribute__((ext_vector_type(2)));
typedef float v8f __attribute__((ext_vector_type(8)));

#define CDIM    128
#define KCHUNK  32
#define XPAD    36    // 32 + 4 : (36*r) mod 64 distinct for r=0..15 -> conflict-free A reads
#define RPITCH  288   // per K-pair row: 2*128 interleaved + 32 pad; 288 mod 64 == 32
                      // -> lanes 0-15 and 16-31 hit disjoint 32-bank halves on b64 reads

// ---------------------------------------------------------------------------
// degree / normalization
// ---------------------------------------------------------------------------
__global__ void k_deg_init(float* __restrict__ deg, int N) {
    int i = blockIdx.x * blockDim.x + threadIdx.x;
    if (i < N) deg[i] = 1.0f;                 // self loop contributes 1
}

__global__ void k_zero_stats(float* __restrict__ s) {
    s[threadIdx.x] = 0.0f;                    // 256 threads: sum[128] + sumsq[128]
}

__global__ void k_deg_accum(const int* __restrict__ dst, float* __restrict__ deg, int E) {
    int e = blockIdx.x * blockDim.x + threadIdx.x;
    if (e < E) atomicAdd(&deg[dst[e]], 1.0f);
}

__global__ void k_dinv(float* __restrict__ deg, int N) {
    int i = blockIdx.x * blockDim.x + threadIdx.x;
    if (i < N) deg[i] = rsqrtf(deg[i]);       // deg >= 1 always (self loop)
}

// ---------------------------------------------------------------------------
// h = x @ W via V_WMMA_F32_16X16X4_F32; epilogue fuses self-loop + bias:
//   h[r,c]   = (x@W)[r,c]
//   out[r,c] = h[r,c] * dinv[r]^2 + b[c]
// W staged K-pair-interleaved: (k,n) -> sw[(k>>1)*RPITCH + 2n + (k&1)]
// so each B fragment is one aligned ds_load_b64. Per k-step we prefetch all
// 8 B fragments into registers, then issue the 8 WMMAs as one dense burst.
// ---------------------------------------------------------------------------
__global__ __launch_bounds__(256) void k_gemm_wmma(
        const float* __restrict__ x, const float* __restrict__ Wm,
        const float* __restrict__ bias, const float* __restrict__ dinv,
        float* __restrict__ h, float* __restrict__ out, int N) {
    __shared__ float sx[128 * XPAD];                 // 128 rows x 32 K (padded)   18KB
    __shared__ float sw[(KCHUNK / 2) * RPITCH];      // 16 K-pairs x 128 cols      18KB

    const int t    = threadIdx.x;
    const int r0   = blockIdx.x * 128;
    const int wave = t >> 5;                  // 0..7 -> 16-row slice
    const int lane = t & 31;
    const int lrow = lane & 15;               // M (A/D) or N (B/D) index
    const int lhi  = lane >> 4;               // K-pair / M+8 selector
    const int mrow = wave * 16;

    v8f acc[8];
#pragma unroll
    for (int i = 0; i < 8; ++i) acc[i] = (v8f){0,0,0,0,0,0,0,0};

    for (int kb = 0; kb < CDIM; kb += KCHUNK) {
        __syncthreads();
        // stage W chunk interleaved: 16 K-pairs x 128 cols = 2048 float2
#pragma unroll
        for (int i = 0; i < 8; ++i) {
            int idx2 = i * 256 + t;           // 0..2047
            int rp   = idx2 >> 7;             // K-pair index (128 float2 per rp)
            int n    = idx2 & 127;
            float w0 = Wm[(size_t)(kb + 2 * rp)     * CDIM + n];
            float w1 = Wm[(size_t)(kb + 2 * rp + 1) * CDIM + n];
            *(float2*)(sw + rp * RPITCH + 2 * n) = make_float2(w0, w1);
        }
        // stage x chunk: rows r0..r0+127, cols kb..kb+31 (1024 float4)
#pragma unroll
        for (int i = 0; i < 4; ++i) {
            int idx4 = i * 256 + t;
            int row  = idx4 >> 3;             // 8 float4 per row
            int col  = (idx4 & 7) << 2;
            int gr   = r0 + row;
            float4 v = make_float4(0.f, 0.f, 0.f, 0.f);
            if (gr < N) v = *(const float4*)(x + (size_t)gr * CDIM + kb + col);
            *(float4*)(sx + row * XPAD + col) = v;
        }
        __syncthreads();

#pragma unroll
        for (int k0 = 0; k0 < KCHUNK; k0 += 4) {
            const int ka = k0 + 2 * lhi;      // even -> 8B-aligned ds_load_b64
            const int rp = (k0 >> 1) + lhi;   // K-pair row for B fragments
            v2f a = *(const v2f*)(sx + (mrow + lrow) * XPAD + ka);
            const float* bptr = sw + rp * RPITCH + 2 * lrow;
            v2f bf[8];
#pragma unroll
            for (int tl = 0; tl < 8; ++tl)    // 4x ds_load_2addr_b64, back-to-back
                bf[tl] = *(const v2f*)(bptr + tl * 32);
#pragma unroll
            for (int tl = 0; tl < 8; ++tl)    // dense WMMA burst
                acc[tl] = __builtin_amdgcn_wmma_f32_16x16x4_f32(
                    false, a, false, bf[tl], (short)0, acc[tl], false, false);
        }
    }

    // epilogue: D element (M = j + 8*lhi, N = lrow) in VGPR j
#pragma unroll
    for (int j = 0; j < 8; ++j) {
        int gr = r0 + mrow + 8 * lhi + j;
        if (gr < N) {
            float d  = dinv[gr];
            float dd = d * d;
#pragma unroll
            for (int tl = 0; tl < 8; ++tl) {
                int   col = tl * 16 + lrow;
                float v   = acc[tl][j];
                h[(size_t)gr * CDIM + col]   = v;
                out[(size_t)gr * CDIM + col] = v * dd + bias[col];
            }
        }
    }
}

// ---------------------------------------------------------------------------
// edge scatter: one wave per edge; lane handles 4 channels (32*4 = 128).
// h and out are L2-resident (2 x 51.2MB << 192MB) so atomics resolve in L2.
// ---------------------------------------------------------------------------
__global__ void k_scatter(const int* __restrict__ src, const int* __restrict__ dst,
                          const float* __restrict__ h, const float* __restrict__ dinv,
                          float* __restrict__ out, int E) {
    int e = blockIdx.x * 8 + (threadIdx.x >> 5);
    if (e >= E) return;
    int lane = threadIdx.x & 31;
    int s = src[e], d = dst[e];
    float w = dinv[s] * dinv[d];
    const float4 hv = *(const float4*)(h + (size_t)s * CDIM + lane * 4);
    float* op = out + (size_t)d * CDIM + lane * 4;
    atomicAdd(op + 0, hv.x * w);
    atomicAdd(op + 1, hv.y * w);
    atomicAdd(op + 2, hv.z * w);
    atomicAdd(op + 3, hv.w * w);
}

// ---------------------------------------------------------------------------
// BatchNorm: per-channel sum / sumsq -> coeffs -> fused scale+shift+ReLU
// ---------------------------------------------------------------------------
__global__ void k_bn_stats(const float* __restrict__ out,
                           float* __restrict__ gsum, float* __restrict__ gsq, int N) {
    const int c    = threadIdx.x & 127;
    const int half = threadIdx.x >> 7;
    float s = 0.f, q = 0.f;
    for (int r = blockIdx.x * 2 + half; r < N; r += gridDim.x * 2) {
        float v = out[(size_t)r * CDIM + c];
        s += v; q += v * v;
    }
    __shared__ float ls[256], lq[256];
    ls[threadIdx.x] = s; lq[threadIdx.x] = q;
    __syncthreads();
    if (threadIdx.x < 128) {
        atomicAdd(&gsum[c], ls[c] + ls[c + 128]);
        atomicAdd(&gsq[c],  lq[c] + lq[c + 128]);
    }
}

__global__ void k_bn_coeff(const float* __restrict__ gsum, const float* __restrict__ gsq,
                           const float* __restrict__ gamma, const float* __restrict__ beta,
                           float* __restrict__ a, float* __restrict__ cc, int N) {
    int c = threadIdx.x;
    if (c >= 128) return;
    float invN = 1.0f / (float)N;
    float mean = gsum[c] * invN;
    float var  = gsq[c] * invN - mean * mean;   // biased variance (ddof=0)
    float sc   = gamma[c] * rsqrtf(var + 1e-5f);
    a[c]  = sc;
    cc[c] = beta[c] - mean * sc;
}

__global__ void k_bn_apply(float* __restrict__ out, const float* __restrict__ a,
                           const float* __restrict__ cc, int total) {
    int i = blockIdx.x * blockDim.x + threadIdx.x;
    if (i >= total) return;
    int c = i & 127;
    float v = out[i] * a[c] + cc[c];
    out[i] = v > 0.f ? v : 0.f;
}

// ---------------------------------------------------------------------------
extern "C" void kernel_launch(void* const* d_in, const int* in_sizes, int n_in,
                              void* d_out, int out_size, void* d_ws, size_t ws_size,
                              hipStream_t stream) {
    const float* x     = (const float*)d_in[0];
    const int*   ei    = (const int*)  d_in[1];   // [2, E] flattened row-major
    const float* Wm    = (const float*)d_in[2];
    const float* bias  = (const float*)d_in[3];
    const float* gamma = (const float*)d_in[4];
    const float* beta  = (const float*)d_in[5];
    float*       out   = (float*)d_out;

    const int N = in_sizes[0] / CDIM;
    const int E = in_sizes[1] / 2;
    const int* srcIdx = ei;
    const int* dstIdx = ei + E;

    // workspace layout (floats): h[N*128] | dinv[N] | sum[128] | sq[128] | a[128] | c[128]
    float* ws   = (float*)d_ws;
    float* h    = ws;
    float* dinv = ws + (size_t)N * CDIM;
    float* gsum = dinv + N;
    float* gsq  = gsum + 128;
    float* abuf = gsq  + 128;
    float* cbuf = abuf + 128;

    k_deg_init <<<(N + 255) / 256, 256, 0, stream>>>(dinv, N);
    k_zero_stats<<<1, 256, 0, stream>>>(gsum);                 // zeroes gsum+gsq
    k_deg_accum<<<(E + 255) / 256, 256, 0, stream>>>(dstIdx, dinv, E);
    k_dinv     <<<(N + 255) / 256, 256, 0, stream>>>(dinv, N);

    k_gemm_wmma<<<(N + 127) / 128, 256, 0, stream>>>(x, Wm, bias, dinv, h, out, N);
    k_scatter  <<<(E + 7) / 8, 256, 0, stream>>>(srcIdx, dstIdx, h, dinv, out, E);

    k_bn_stats <<<512, 256, 0, stream>>>(out, gsum, gsq, N);
    k_bn_coeff <<<1, 128, 0, stream>>>(gsum, gsq, gamma, beta, abuf, cbuf, N);
    const int total = N * CDIM;
    k_bn_apply <<<(total + 255) / 256, 256, 0, stream>>>(out, abuf, cbuf, total);
}